// GraphConvolution_31439160607139
// MI455X (gfx1250) — compile-verified
//
#include <hip/hip_runtime.h>
#include <math.h>

// ---------------------------------------------------------------------------
// Problem constants (match reference)
// ---------------------------------------------------------------------------
#define N_NODES 25000
#define N_EDGES 400000
#define MUL0    64
#define MUL1    32
#define DIM_IN  160       // 64 + 3*32
#define NB      10
#define NH      100
#define WN      192       // 64+64+32+32
#define MIDW    384       // 64 + 32 + 3*64 + 3*32

__device__ __constant__ float C_S        = 0.3826834323650898f;  // sin(pi/8)
__device__ __constant__ float C_X        = 0.9238795325112867f;  // cos(pi/8)
#define INV_SQRT3   0.57735026918962576f
#define INV_SQRT64  0.125f
#define INV_SQRT32  0.17677669529663687f
#define INV_SQRT10  0.31622776601683794f
#define INV_SQRT100 0.1f
#define INV_SQRT96  0.10206207261596577f

typedef __attribute__((ext_vector_type(2))) float v2f;
typedef __attribute__((ext_vector_type(8))) float v8f;

__device__ __forceinline__ v8f v8f_zero() {
  v8f z = {0.f, 0.f, 0.f, 0.f, 0.f, 0.f, 0.f, 0.f};
  return z;
}

// D = A(16x4,f32) * B(4x16,f32) + C  -- native fp32 WMMA on gfx1250
__device__ __forceinline__ v8f wmma4(v2f a, v2f b, v8f c) {
  return __builtin_amdgcn_wmma_f32_16x16x4_f32(
      /*neg_a=*/false, a, /*neg_b=*/false, b,
      /*c_mod=*/(short)0, c, /*reuse_a=*/false, /*reuse_b=*/false);
}

// B fragment: B is (K x n) row-major, leading dim ld. Tile rows k0..k0+3,
// cols n0..n0+15.  lane = N column; lane halves split K.
__device__ __forceinline__ v2f ld_b_frag(const float* __restrict__ B, int ld,
                                         int k0, int n0) {
  const int lane = threadIdx.x & 31;
  const int n = n0 + (lane & 15);
  const int k = k0 + 2 * (lane >> 4);
  v2f b;
  b.x = B[(size_t)k * ld + n];
  b.y = B[(size_t)(k + 1) * ld + n];
  return b;
}

// A fragment, strided columns: logical K index k -> column (colbase+k*cstride)
__device__ __forceinline__ v2f ld_a_strided(const float* __restrict__ A, int r,
                                            int ld, int colbase, int cstride,
                                            int k0) {
  const int lane = threadIdx.x & 31;
  const int k = k0 + 2 * (lane >> 4);
  const float* p = A + (size_t)r * ld + colbase + (size_t)k * cstride;
  v2f a;
  a.x = p[0];
  a.y = p[cstride];
  return a;
}

// A fragment, contiguous columns: single 8-byte load (colbase even, k even,
// row stride a multiple of 8B -> aligned)
__device__ __forceinline__ v2f ld_a_contig(const float* __restrict__ A, int r,
                                           int ld, int colbase, int k0) {
  const int lane = threadIdx.x & 31;
  const int k = k0 + 2 * (lane >> 4);
  return *(const v2f*)(A + (size_t)r * ld + colbase + k);
}

// ---------------------------------------------------------------------------
// Kernel 1: node input transforms.
//   f0 = x0@W_in0 * attr * inv_sqrt64 * inv_sqrt_deg            -> ws
//   f1 = x1@W_in1 * attr * inv_sqrt32 * inv_sqrt_deg            -> ws ([w][i])
//   out = C_S * (x@W_mask)  (seeds the final output)
// One 16-row tile per wave, 2 waves per block.
// ---------------------------------------------------------------------------
__global__ __launch_bounds__(64) void node_in_kernel(
    const float* __restrict__ node_input, const float* __restrict__ node_attr,
    const float* __restrict__ node_deg, const float* __restrict__ W_in0,
    const float* __restrict__ W_in1, const float* __restrict__ W_mask0,
    const float* __restrict__ W_mask1, float* __restrict__ f0,
    float* __restrict__ f1, float* __restrict__ out) {
  const int lane = threadIdx.x & 31;
  const int wv = threadIdx.x >> 5;
  const int tile = blockIdx.x * 2 + wv;
  const int row0 = tile * 16;
  if (row0 >= N_NODES) return;
  if (threadIdx.x == 0) {
    __builtin_prefetch(W_in0, 0, 3);
    __builtin_prefetch(W_mask0, 0, 3);
  }
  int r = row0 + (lane & 15);
  if (r >= N_NODES) r = N_NODES - 1;   // clamped load row; stores are guarded

  const int hi = lane >> 4, n = lane & 15;

  // ---- scalar channels: (16x64)@(64x64) for W_in0 and W_mask0 ----
  {
    v8f af[4], am[4];
#pragma unroll
    for (int t = 0; t < 4; ++t) { af[t] = v8f_zero(); am[t] = v8f_zero(); }
    for (int kk = 0; kk < 16; ++kk) {
      v2f a = ld_a_contig(node_input, r, DIM_IN, 0, kk * 4);
#pragma unroll
      for (int t = 0; t < 4; ++t) {
        af[t] = wmma4(a, ld_b_frag(W_in0, MUL0, kk * 4, t * 16), af[t]);
        am[t] = wmma4(a, ld_b_frag(W_mask0, MUL0, kk * 4, t * 16), am[t]);
      }
    }
#pragma unroll
    for (int t = 0; t < 4; ++t) {
#pragma unroll
      for (int v = 0; v < 8; ++v) {
        const int rr = row0 + v + 8 * hi;
        if (rr < N_NODES) {
          const float attr = node_attr[rr];
          const float isd = rsqrtf(node_deg[rr]);
          const int c = t * 16 + n;
          f0[(size_t)rr * MUL0 + c] = af[t][v] * attr * INV_SQRT64 * isd;
          out[(size_t)rr * DIM_IN + c] = C_S * (am[t][v] * attr * INV_SQRT64);
        }
      }
    }
  }

  // ---- vector channels: per component i, (16x32)@(32x32) ----
#pragma unroll
  for (int i = 0; i < 3; ++i) {
    v8f bf[2], bm[2];
#pragma unroll
    for (int t = 0; t < 2; ++t) { bf[t] = v8f_zero(); bm[t] = v8f_zero(); }
    for (int kk = 0; kk < 8; ++kk) {
      v2f a = ld_a_strided(node_input, r, DIM_IN, MUL0 + i, 3, kk * 4);
#pragma unroll
      for (int t = 0; t < 2; ++t) {
        bf[t] = wmma4(a, ld_b_frag(W_in1, MUL1, kk * 4, t * 16), bf[t]);
        bm[t] = wmma4(a, ld_b_frag(W_mask1, MUL1, kk * 4, t * 16), bm[t]);
      }
    }
#pragma unroll
    for (int t = 0; t < 2; ++t) {
#pragma unroll
      for (int v = 0; v < 8; ++v) {
        const int rr = row0 + v + 8 * hi;
        if (rr < N_NODES) {
          const float attr = node_attr[rr];
          const float isd = rsqrtf(node_deg[rr]);
          const int w = t * 16 + n;
          f1[(size_t)rr * (3 * MUL1) + 3 * w + i] =
              bf[t][v] * attr * INV_SQRT32 * isd;
          out[(size_t)rr * DIM_IN + MUL0 + 3 * w + i] =
              C_S * (bm[t][v] * attr * INV_SQRT32);
        }
      }
    }
  }
}

// ---------------------------------------------------------------------------
// Kernel 2: fused edge MLP + tensor-product features + scatter-add.
// 4 waves / block, 16 edges / wave.
//   phase 1 (regs):  each lane computes exactly the h values its A-fragments
//                    need: h[m=lane&15][k] for k = kk*4 + 2*(lane>>4)+{0,1},
//                    kk = 0..24.  No LDS, no barrier for h.
//   phase 2 (WMMA):  ew = h @ W_mlp2 * inv_sqrt100  (16x192, K=100),
//                    D fragments transposed through LDS.
//   phase 3 (scalar): build 384 edge features, atomicAdd into node_mid
//                    (node_mid is L2-resident: 38 MB < 192 MB L2).
// ---------------------------------------------------------------------------
#define K2_WAVES 4
__global__ __launch_bounds__(32 * K2_WAVES) void edge_kernel(
    const float* __restrict__ elen, const float* __restrict__ W_mlp1,
    const float* __restrict__ W_mlp2, const int* __restrict__ edge_src,
    const int* __restrict__ edge_dst, const float* __restrict__ edge_attr,
    const float* __restrict__ f0, const float* __restrict__ f1,
    float* __restrict__ node_mid) {
  __shared__ float lds_ew[K2_WAVES][16 * WN];
  const int lane = threadIdx.x & 31;
  const int wv = threadIdx.x >> 5;
  const int e0 = blockIdx.x * (16 * K2_WAVES) + wv * 16;
  const int rm = lane & 15;
  const int hi = lane >> 4;

  if (threadIdx.x == 0) {
    __builtin_prefetch(W_mlp2, 0, 3);
    __builtin_prefetch(W_mlp2 + 32, 0, 3);
    __builtin_prefetch(W_mlp1, 0, 3);
  }

  // ---- phase 1: hidden layer + SiLU straight into A-fragment registers ----
  int ge_a = e0 + rm;
  if (ge_a >= N_EDGES) ge_a = N_EDGES - 1;
  float el[NB];
#pragma unroll
  for (int t = 0; t < NB; ++t) el[t] = elen[(size_t)ge_a * NB + t];

  v2f hfrag[25];
#pragma unroll
  for (int kk = 0; kk < 25; ++kk) {
    const int j0 = kk * 4 + 2 * hi;
    float s0 = 0.f, s1 = 0.f;
#pragma unroll
    for (int t = 0; t < NB; ++t) {
      s0 += el[t] * W_mlp1[t * NH + j0];
      s1 += el[t] * W_mlp1[t * NH + j0 + 1];
    }
    s0 *= INV_SQRT10;
    s1 *= INV_SQRT10;
    hfrag[kk].x = s0 / (1.f + expf(-s0));   // silu
    hfrag[kk].y = s1 / (1.f + expf(-s1));
  }

  // ---- phase 2: ew = h @ W_mlp2 (16x192, K=100) via fp32 WMMA ----
  {
    v8f acc[12];
#pragma unroll
    for (int t = 0; t < 12; ++t) acc[t] = v8f_zero();
    for (int kk = 0; kk < 25; ++kk) {
#pragma unroll
      for (int t = 0; t < 12; ++t)
        acc[t] = wmma4(hfrag[kk], ld_b_frag(W_mlp2, WN, kk * 4, t * 16),
                       acc[t]);
    }
#pragma unroll
    for (int t = 0; t < 12; ++t)
#pragma unroll
      for (int v = 0; v < 8; ++v)
        lds_ew[wv][(v + 8 * hi) * WN + t * 16 + rm] = acc[t][v] * INV_SQRT100;
  }
  __syncthreads();

  // ---- phase 3: tensor-product features + scatter ----
  for (int idx = lane; idx < 16 * MIDW; idx += 32) {
    const int e = idx / MIDW;
    const int col = idx - e * MIDW;
    const int ge = e0 + e;
    if (ge >= N_EDGES) continue;
    const int src = edge_src[ge];
    const int dst = edge_dst[ge];
    const float* ew = &lds_ew[wv][e * WN];
    const float ea0 = edge_attr[(size_t)ge * 4];
    float val;
    if (col < MUL0) {                       // mid0a: w_i0 * xs0 * ea0
      val = ew[col] * f0[(size_t)src * MUL0 + col] * ea0;
    } else if (col < MUL0 + MUL1) {         // mid0b: w_i3 * <xs1,ea1> / sqrt3
      const int u = col - MUL0;
      float d = 0.f;
#pragma unroll
      for (int i = 0; i < 3; ++i)
        d += f1[(size_t)src * (3 * MUL1) + 3 * u + i] *
             edge_attr[(size_t)ge * 4 + 1 + i];
      val = ew[160 + u] * d * INV_SQRT3;
    } else if (col < MUL0 + MUL1 + 3 * MUL0) {  // mid1a: w_i1 * xs0 * ea1[i]
      const int t = col - (MUL0 + MUL1);
      const int u = t / 3;
      const int i = t - 3 * u;
      val = ew[64 + u] * f0[(size_t)src * MUL0 + u] *
            edge_attr[(size_t)ge * 4 + 1 + i];
    } else {                                // mid1b: w_i2 * xs1 * ea0
      const int t = col - (MUL0 + MUL1 + 3 * MUL0);
      const int u = t / 3;
      val = ew[128 + u] * f1[(size_t)src * (3 * MUL1) + t] * ea0;
    }
    atomicAdd(&node_mid[(size_t)dst * MIDW + col], val);
  }
}

// ---------------------------------------------------------------------------
// Kernel 3: node output transforms.
//   o0 = (g0a@W_out0a + g0b@W_out0b) * attr * inv_sqrt96
//   o1 = (g1a@W_out1a + g1b@W_out1b) * attr * inv_sqrt96   (per component)
//   out += C_X * o          (A rows pre-scaled by inv_sqrt_deg)
// ---------------------------------------------------------------------------
__global__ __launch_bounds__(64) void node_out_kernel(
    const float* __restrict__ node_mid, const float* __restrict__ node_attr,
    const float* __restrict__ node_deg, const float* __restrict__ W_out0a,
    const float* __restrict__ W_out0b, const float* __restrict__ W_out1a,
    const float* __restrict__ W_out1b, float* __restrict__ out) {
  const int lane = threadIdx.x & 31;
  const int wv = threadIdx.x >> 5;
  const int tile = blockIdx.x * 2 + wv;
  const int row0 = tile * 16;
  if (row0 >= N_NODES) return;
  if (threadIdx.x == 0) {
    __builtin_prefetch(W_out0a, 0, 3);
    __builtin_prefetch(W_out1a, 0, 3);
  }
  int r = row0 + (lane & 15);
  if (r >= N_NODES) r = N_NODES - 1;
  const float isd_a = rsqrtf(node_deg[r]);   // scales A-matrix rows
  const int hi = lane >> 4, n = lane & 15;

  // ---- o0: (16x64)@(64x64) + (16x32)@(32x64) ----
  {
    v8f acc[4];
#pragma unroll
    for (int t = 0; t < 4; ++t) acc[t] = v8f_zero();
    for (int kk = 0; kk < 16; ++kk) {   // g0a, K=64
      v2f a = ld_a_contig(node_mid, r, MIDW, 0, kk * 4);
      a.x *= isd_a; a.y *= isd_a;
#pragma unroll
      for (int t = 0; t < 4; ++t)
        acc[t] = wmma4(a, ld_b_frag(W_out0a, MUL0, kk * 4, t * 16), acc[t]);
    }
    for (int kk = 0; kk < 8; ++kk) {    // g0b, K=32
      v2f a = ld_a_contig(node_mid, r, MIDW, MUL0, kk * 4);
      a.x *= isd_a; a.y *= isd_a;
#pragma unroll
      for (int t = 0; t < 4; ++t)
        acc[t] = wmma4(a, ld_b_frag(W_out0b, MUL0, kk * 4, t * 16), acc[t]);
    }
#pragma unroll
    for (int t = 0; t < 4; ++t)
#pragma unroll
      for (int v = 0; v < 8; ++v) {
        const int rr = row0 + v + 8 * hi;
        if (rr < N_NODES) {
          const float attr = node_attr[rr];
          const size_t o = (size_t)rr * DIM_IN + t * 16 + n;
          out[o] += C_X * acc[t][v] * attr * INV_SQRT96;
        }
      }
  }

  // ---- o1 per component: (16x64)@(64x32) + (16x32)@(32x32) ----
#pragma unroll
  for (int i = 0; i < 3; ++i) {
    v8f acc[2];
#pragma unroll
    for (int t = 0; t < 2; ++t) acc[t] = v8f_zero();
    for (int kk = 0; kk < 16; ++kk) {   // g1a, K=64, stride-3 columns
      v2f a = ld_a_strided(node_mid, r, MIDW, MUL0 + MUL1 + i, 3, kk * 4);
      a.x *= isd_a; a.y *= isd_a;
#pragma unroll
      for (int t = 0; t < 2; ++t)
        acc[t] = wmma4(a, ld_b_frag(W_out1a, MUL1, kk * 4, t * 16), acc[t]);
    }
    for (int kk = 0; kk < 8; ++kk) {    // g1b, K=32
      v2f a = ld_a_strided(node_mid, r, MIDW, MUL0 + MUL1 + 3 * MUL0 + i, 3,
                           kk * 4);
      a.x *= isd_a; a.y *= isd_a;
#pragma unroll
      for (int t = 0; t < 2; ++t)
        acc[t] = wmma4(a, ld_b_frag(W_out1b, MUL1, kk * 4, t * 16), acc[t]);
    }
#pragma unroll
    for (int t = 0; t < 2; ++t)
#pragma unroll
      for (int v = 0; v < 8; ++v) {
        const int rr = row0 + v + 8 * hi;
        if (rr < N_NODES) {
          const float attr = node_attr[rr];
          const int w = t * 16 + n;
          const size_t o = (size_t)rr * DIM_IN + MUL0 + 3 * w + i;
          out[o] += C_X * acc[t][v] * attr * INV_SQRT96;
        }
      }
  }
}

// ---------------------------------------------------------------------------
extern "C" void kernel_launch(void* const* d_in, const int* in_sizes, int n_in,
                              void* d_out, int out_size, void* d_ws,
                              size_t ws_size, hipStream_t stream) {
  const float* node_input = (const float*)d_in[0];
  const float* node_attr  = (const float*)d_in[1];
  const float* node_deg   = (const float*)d_in[2];
  const int*   edge_src   = (const int*)d_in[3];
  const int*   edge_dst   = (const int*)d_in[4];
  const float* edge_attr  = (const float*)d_in[5];
  const float* elen       = (const float*)d_in[6];
  const float* W_in0      = (const float*)d_in[7];
  const float* W_in1      = (const float*)d_in[8];
  const float* W_mask0    = (const float*)d_in[9];
  const float* W_mask1    = (const float*)d_in[10];
  const float* W_out0a    = (const float*)d_in[11];
  const float* W_out0b    = (const float*)d_in[12];
  const float* W_out1a    = (const float*)d_in[13];
  const float* W_out1b    = (const float*)d_in[14];
  const float* W_mlp1     = (const float*)d_in[15];
  const float* W_mlp2     = (const float*)d_in[16];
  float* out = (float*)d_out;

  // Workspace layout
  float* f0 = (float*)d_ws;                              // N x 64
  float* f1 = f0 + (size_t)N_NODES * MUL0;               // N x 96
  float* node_mid = f1 + (size_t)N_NODES * 3 * MUL1;     // N x 384

  hipMemsetAsync(node_mid, 0, (size_t)N_NODES * MIDW * sizeof(float), stream);

  const int tiles = (N_NODES + 15) / 16;         // 1563
  dim3 ngrid((tiles + 1) / 2), nblk(64);

  node_in_kernel<<<ngrid, nblk, 0, stream>>>(node_input, node_attr, node_deg,
                                             W_in0, W_in1, W_mask0, W_mask1,
                                             f0, f1, out);

  dim3 egrid((N_EDGES + 16 * K2_WAVES - 1) / (16 * K2_WAVES)),
      eblk(32 * K2_WAVES);
  edge_kernel<<<egrid, eblk, 0, stream>>>(elen, W_mlp1, W_mlp2, edge_src,
                                          edge_dst, edge_attr, f0, f1,
                                          node_mid);

  node_out_kernel<<<ngrid, nblk, 0, stream>>>(node_mid, node_attr, node_deg,
                                              W_out0a, W_out0b, W_out1a,
                                              W_out1b, out);
}